// MLAVL_88433376624848
// MI455X (gfx1250) — compile-verified
//
#include <hip/hip_runtime.h>

// ---------------------------------------------------------------------------
// MLAVL dual-SAGE graph conv, rewritten as dense fp32 WMMA GEMMs (gfx1250).
//
// Key identity: the edge set is (complete visual)+(complete text)+(text->vis),
// so mean-aggregation is   agg_i = (S_class(i) - x_i) * alpha_i
// with S = column sum over the node class block and alpha in {1/177, 1/49}.
// Each SAGEConv layer y = lin_l(mean_agg) + lin_r(x) therefore becomes:
//     P = x @ Wl ; Q = x @ Wr ; y_i = Q_i + alpha_i*(S_cls @ Wl - P_i) + b
// -> two dense GEMMs per layer on V_WMMA_F32_16X16X4_F32 (exact fp32 path),
//    plus tiny column-sum / GEMV helper kernels.
// ---------------------------------------------------------------------------

#define TSEQ   128           // visual nodes
#define NTEXT  50            // text nodes
#define NNODES 178           // TSEQ + NTEXT
#define NPAD   192           // 12 row-tiles of 16
#define NRT    12            // row tiles
#define HID    512
#define GHID   256
#define NBB    16            // 2 branches * 8 batch

#define LDA1   516           // LDS row stride (floats): 4*row+k hits 32 banks
#define LDA2   260

typedef float v2f __attribute__((ext_vector_type(2)));
typedef float v8f __attribute__((ext_vector_type(8)));

__device__ __forceinline__ float x_elem(const float* __restrict__ vis,
                                        const float* __restrict__ txt,
                                        int b, int g, int k) {
  if (g < TSEQ)   return vis[(b * TSEQ + g) * HID + k];
  if (g < NNODES) return txt[(b * NTEXT + (g - TSEQ)) * HID + k];
  return 0.0f;                                    // pad rows
}

// ---- pass 1: column sums of the input node features ------------------------
__global__ __launch_bounds__(HID)
void colsum_x_kernel(const float* __restrict__ ev, const float* __restrict__ ea,
                     const float* __restrict__ tv, const float* __restrict__ ta,
                     float* __restrict__ s1) {
  const int bb = blockIdx.x, br = bb >> 3, b = bb & 7;
  const float* vis = br ? ea : ev;
  const float* txt = br ? ta : tv;
  const int k = threadIdx.x;                      // 0..511, coalesced
  float sv = 0.f, st = 0.f;
  for (int r = 0; r < TSEQ; ++r)  sv += vis[(b * TSEQ + r) * HID + k];
  for (int r = 0; r < NTEXT; ++r) st += txt[(b * NTEXT + r) * HID + k];
  s1[(bb * 2 + 0) * HID + k] = sv + st;           // class 0: all nodes
  s1[(bb * 2 + 1) * HID + k] = st;                // class 1: text nodes
}

// ---- pass 2: t1[bb][cls][n] = s1[bb][cls] @ W1l ---------------------------
__global__ __launch_bounds__(GHID)
void gemv1_kernel(const float* __restrict__ Wv1l, const float* __restrict__ Wa1l,
                  const float* __restrict__ s1, float* __restrict__ t1) {
  const int bb = blockIdx.x, br = bb >> 3;
  const float* W  = br ? Wa1l : Wv1l;             // [512,256] row-major
  const int n = threadIdx.x;                      // 0..255
  const float* s0 = &s1[(bb * 2 + 0) * HID];
  const float* sT = &s1[(bb * 2 + 1) * HID];
  float a0 = 0.f, a1 = 0.f;
  for (int k = 0; k < HID; ++k) {
    const float w = W[k * GHID + n];
    a0 = fmaf(s0[k], w, a0);
    a1 = fmaf(sT[k], w, a1);
  }
  t1[(bb * 2 + 0) * GHID + n] = a0;
  t1[(bb * 2 + 1) * GHID + n] = a1;
}

// ---- pass 3: layer-1 GEMMs on WMMA + fused SAGE/ReLU epilogue -------------
__global__ __launch_bounds__(512)
void sage1_kernel(const float* __restrict__ ev, const float* __restrict__ ea,
                  const float* __restrict__ tv, const float* __restrict__ ta,
                  const float* __restrict__ Wv1l, const float* __restrict__ bv1l,
                  const float* __restrict__ Wv1r,
                  const float* __restrict__ Wa1l, const float* __restrict__ ba1l,
                  const float* __restrict__ Wa1r,
                  const float* __restrict__ t1, float* __restrict__ h) {
  __shared__ float Alds[16 * LDA1];               // 33 KB of the 320 KB WGP LDS
  const int rowTile = blockIdx.x;                 // 0..11
  const int bb = blockIdx.y, br = bb >> 3, b = bb & 7;
  const float* vis  = br ? ea   : ev;
  const float* txt  = br ? ta   : tv;
  const float* Wl   = br ? Wa1l : Wv1l;
  const float* Wr   = br ? Wa1r : Wv1r;
  const float* bias = br ? ba1l : bv1l;

  // Stage the 16x512 A tile once; shared by all 16 waves. Uniform trip count.
  for (int idx = threadIdx.x; idx < 16 * HID; idx += 512) {
    const int row = idx >> 9, k = idx & (HID - 1);
    Alds[row * LDA1 + k] = x_elem(vis, txt, b, rowTile * 16 + row, k);
  }
  __syncthreads();

  const int wave = threadIdx.x >> 5;              // col tile 0..15
  const int lane = threadIdx.x & 31;
  const int half = lane >> 4, lm = lane & 15;
  const int col0 = wave * 16;

  v8f accP = {};                                  // x @ Wl
  v8f accQ = {};                                  // x @ Wr
  #pragma unroll 4
  for (int kk = 0; kk < HID; kk += 4) {
    const int ka = kk + 2 * half;                 // A: lanes 16..31 hold K+2
    v2f a, bl_, br_;
    a.x   = Alds[lm * LDA1 + ka];
    a.y   = Alds[lm * LDA1 + ka + 1];
    bl_.x = Wl[(ka    ) * GHID + col0 + lm];      // B: row-per-VGPR layout
    bl_.y = Wl[(ka + 1) * GHID + col0 + lm];
    br_.x = Wr[(ka    ) * GHID + col0 + lm];
    br_.y = Wr[(ka + 1) * GHID + col0 + lm];
    accP = __builtin_amdgcn_wmma_f32_16x16x4_f32(false, a, false, bl_,
                                                 (short)0, accP, false, false);
    accQ = __builtin_amdgcn_wmma_f32_16x16x4_f32(false, a, false, br_,
                                                 (short)0, accQ, false, false);
  }

  // Row tiles 0..7 are all-visual, 8..11 all-text (128 = 8*16): uniform class.
  const int   cls   = (rowTile >= 8) ? 1 : 0;
  const float alpha = cls ? (1.0f / 49.0f) : (1.0f / 177.0f);
  const int   n     = col0 + lm;
  const float sumP  = t1[(bb * 2 + cls) * GHID + n];
  const float bn    = bias[n];
  #pragma unroll
  for (int v = 0; v < 8; ++v) {                   // D: M = v + 8*half
    const int g = rowTile * 16 + v + 8 * half;
    float val = accQ[v] + alpha * (sumP - accP[v]) + bn;
    val = fmaxf(val, 0.0f);                       // ReLU
    if (g >= NNODES) val = 0.0f;                  // keep pad rows clean
    h[(bb * NPAD + g) * GHID + n] = val;
  }
}

// ---- pass 4: column sums of h ---------------------------------------------
__global__ __launch_bounds__(GHID)
void colsum_h_kernel(const float* __restrict__ h, float* __restrict__ s2) {
  const int bb = blockIdx.x, n = threadIdx.x;     // 0..255
  const float* hb = &h[bb * NPAD * GHID];
  float sv = 0.f, st = 0.f;
  for (int r = 0; r < TSEQ; ++r)        sv += hb[r * GHID + n];
  for (int r = TSEQ; r < NNODES; ++r)   st += hb[r * GHID + n];
  s2[(bb * 2 + 0) * GHID + n] = sv + st;
  s2[(bb * 2 + 1) * GHID + n] = st;
}

// ---- pass 5: t2 = s2 @ W2l -------------------------------------------------
__global__ __launch_bounds__(HID)
void gemv2_kernel(const float* __restrict__ Wv2l, const float* __restrict__ Wa2l,
                  const float* __restrict__ s2, float* __restrict__ t2) {
  const int bb = blockIdx.x, br = bb >> 3;
  const float* W = br ? Wa2l : Wv2l;              // [256,512]
  const int n = threadIdx.x;                      // 0..511
  const float* s0 = &s2[(bb * 2 + 0) * GHID];
  const float* sT = &s2[(bb * 2 + 1) * GHID];
  float a0 = 0.f, a1 = 0.f;
  for (int k = 0; k < GHID; ++k) {
    const float w = W[k * HID + n];
    a0 = fmaf(s0[k], w, a0);
    a1 = fmaf(sT[k], w, a1);
  }
  t2[(bb * 2 + 0) * HID + n] = a0;
  t2[(bb * 2 + 1) * HID + n] = a1;
}

// ---- pass 6: layer-2 GEMMs + epilogue -> d_out -----------------------------
__global__ __launch_bounds__(1024)
void sage2_kernel(const float* __restrict__ h,
                  const float* __restrict__ Wv2l, const float* __restrict__ bv2l,
                  const float* __restrict__ Wv2r,
                  const float* __restrict__ Wa2l, const float* __restrict__ ba2l,
                  const float* __restrict__ Wa2r,
                  const float* __restrict__ t2, float* __restrict__ out) {
  __shared__ float Alds[16 * LDA2];               // 16.6 KB
  const int rowTile = blockIdx.x;
  const int bb = blockIdx.y, br = bb >> 3;
  const float* Wl   = br ? Wa2l : Wv2l;
  const float* Wr   = br ? Wa2r : Wv2r;
  const float* bias = br ? ba2l : bv2l;
  const float* hb   = &h[(bb * NPAD + rowTile * 16) * GHID];

  for (int idx = threadIdx.x; idx < 16 * GHID; idx += 1024) {
    const int row = idx >> 8, k = idx & (GHID - 1);
    Alds[row * LDA2 + k] = hb[row * GHID + k];
  }
  __syncthreads();

  const int wave = threadIdx.x >> 5;              // 32 waves -> 512 cols
  const int lane = threadIdx.x & 31;
  const int half = lane >> 4, lm = lane & 15;
  const int col0 = wave * 16;

  v8f accP = {};
  v8f accQ = {};
  #pragma unroll 4
  for (int kk = 0; kk < GHID; kk += 4) {
    const int ka = kk + 2 * half;
    v2f a, bl_, br_;
    a.x   = Alds[lm * LDA2 + ka];
    a.y   = Alds[lm * LDA2 + ka + 1];
    bl_.x = Wl[(ka    ) * HID + col0 + lm];
    bl_.y = Wl[(ka + 1) * HID + col0 + lm];
    br_.x = Wr[(ka    ) * HID + col0 + lm];
    br_.y = Wr[(ka + 1) * HID + col0 + lm];
    accP = __builtin_amdgcn_wmma_f32_16x16x4_f32(false, a, false, bl_,
                                                 (short)0, accP, false, false);
    accQ = __builtin_amdgcn_wmma_f32_16x16x4_f32(false, a, false, br_,
                                                 (short)0, accQ, false, false);
  }

  const int   cls   = (rowTile >= 8) ? 1 : 0;
  const float alpha = cls ? (1.0f / 49.0f) : (1.0f / 177.0f);
  const int   n     = col0 + lm;
  const float sumP  = t2[(bb * 2 + cls) * HID + n];
  const float bn    = bias[n];
  #pragma unroll
  for (int v = 0; v < 8; ++v) {
    const int g = rowTile * 16 + v + 8 * half;
    if (g < NNODES)                               // out is [2,8,178,512]
      out[(bb * NNODES + g) * HID + n] = accQ[v] + alpha * (sumP - accP[v]) + bn;
  }
}

// ---------------------------------------------------------------------------
extern "C" void kernel_launch(void* const* d_in, const int* in_sizes, int n_in,
                              void* d_out, int out_size, void* d_ws, size_t ws_size,
                              hipStream_t stream) {
  (void)in_sizes; (void)n_in; (void)out_size; (void)ws_size;
  const float* ev   = (const float*)d_in[0];
  const float* ea   = (const float*)d_in[1];
  const float* tv   = (const float*)d_in[2];
  const float* ta   = (const float*)d_in[3];
  const float* Wv1l = (const float*)d_in[4];
  const float* bv1l = (const float*)d_in[5];
  const float* Wv1r = (const float*)d_in[6];
  const float* Wv2l = (const float*)d_in[7];
  const float* bv2l = (const float*)d_in[8];
  const float* Wv2r = (const float*)d_in[9];
  const float* Wa1l = (const float*)d_in[10];
  const float* ba1l = (const float*)d_in[11];
  const float* Wa1r = (const float*)d_in[12];
  const float* Wa2l = (const float*)d_in[13];
  const float* ba2l = (const float*)d_in[14];
  const float* Wa2r = (const float*)d_in[15];
  // d_in[16]/d_in[17] (src/dst edge lists) are implied by the dense structure.
  float* outp = (float*)d_out;

  // workspace layout (floats): h | s1 | t1 | s2 | t2  (~3.2 MB total)
  float* h  = (float*)d_ws;
  float* s1 = h  + (size_t)NBB * NPAD * GHID;
  float* t1 = s1 + (size_t)NBB * 2 * HID;
  float* s2 = t1 + (size_t)NBB * 2 * GHID;
  float* t2 = s2 + (size_t)NBB * 2 * GHID;

  colsum_x_kernel<<<NBB, HID, 0, stream>>>(ev, ea, tv, ta, s1);
  gemv1_kernel  <<<NBB, GHID, 0, stream>>>(Wv1l, Wa1l, s1, t1);
  sage1_kernel  <<<dim3(NRT, NBB), 512, 0, stream>>>(ev, ea, tv, ta,
                                                     Wv1l, bv1l, Wv1r,
                                                     Wa1l, ba1l, Wa1r, t1, h);
  colsum_h_kernel<<<NBB, GHID, 0, stream>>>(h, s2);
  gemv2_kernel  <<<NBB, HID, 0, stream>>>(Wv2l, Wa2l, s2, t2);
  sage2_kernel  <<<dim3(NRT, NBB), 1024, 0, stream>>>(h,
                                                      Wv2l, bv2l, Wv2r,
                                                      Wa2l, ba2l, Wa2r, t2, outp);
}